// OcnnResNet3D_14937896256229
// MI455X (gfx1250) — compile-verified
//
#include <hip/hip_runtime.h>
#include <hip/hip_bf16.h>

// ---------------------------------------------------------------------------
// OcnnResNet3D stem + resblock on MI455X (gfx1250, wave32, WMMA).
// fp32 path using V_WMMA_F32_16X16X4_F32 (D = A[16x4] * B[4x16] + C[16x16]).
// Weights pair-interleaved in LDS so each B fragment is one ds_load_b64.
// Gather staging uses GLOBAL_LOAD_ASYNC_TO_LDS_B64 when the toolchain has it.
// ---------------------------------------------------------------------------

#define N0_ 1000000
#define N1_ 300000
#define N2_ 80000
#define N3_ 30000
#define CIN_ 4
#define CC_ 64
#define KK_ 27
#define BN_EPS 1e-5f

typedef __attribute__((ext_vector_type(2))) float v2f;
typedef __attribute__((ext_vector_type(8))) float v8f;
typedef __attribute__((ext_vector_type(2))) int v2i;

#if defined(__has_builtin)
#if __has_builtin(__builtin_amdgcn_global_load_async_to_lds_b64) && \
    __has_builtin(__builtin_amdgcn_s_wait_asynccnt)
#define USE_ASYNC_LDS 1
#endif
#endif

__device__ __forceinline__ v8f wmma_f32_k4(v2f a, v2f b, v8f c) {
  // (neg_a, A, neg_b, B, c_mod, C, reuse_a, reuse_b)
  return __builtin_amdgcn_wmma_f32_16x16x4_f32(
      false, a, false, b, (short)0, c, false, false);
}

// ---------------------------------------------------------------------------
// Stem conv: out[n, co] = sum_k sum_ci x[neigh[n,k], ci] * W[k, ci, co]
// M = N1 (tiles of 16 nodes per wave), Ktot = 27*4, N = 64.
// ---------------------------------------------------------------------------
#define STEM_WAVES 4
__global__ __launch_bounds__(32 * STEM_WAVES)
void stem_conv_wmma(const float* __restrict__ x,
                    const int* __restrict__ neigh,
                    const float* __restrict__ W,  // [27][4][64]
                    float* __restrict__ out) {    // [N1][64]
  // interleaved: wlds[k*256 + pair*128 + col*2 + parity], row = 2*pair+parity
  __shared__ float wlds[KK_ * CIN_ * CC_];          // 6912 floats
  __shared__ int nlds[STEM_WAVES][16 * KK_];        // 432 ints / wave

  const int tid  = threadIdx.x;
  const int wave = tid >> 5;
  const int lane = tid & 31;
  const int m    = lane & 15;
  const int hi   = lane >> 4;

  for (int i = tid; i < KK_ * 2 * CC_; i += blockDim.x) {
    const int k = i >> 7;
    const int r = (i >> 6) & 1;   // row pair within k
    const int c = i & 63;
    v2f v;
    v.x = W[k * 256 + (2 * r) * CC_ + c];
    v.y = W[k * 256 + (2 * r + 1) * CC_ + c];
    *(v2f*)&wlds[k * 256 + r * 128 + c * 2] = v;
  }

  const int numTiles = N1_ / 16;
  const int tile = blockIdx.x * STEM_WAVES + wave;
  const bool active = tile < numTiles;
  if (active) {
    const int base = tile * 16 * KK_;
    for (int i = lane; i < 16 * KK_; i += 32) nlds[wave][i] = neigh[base + i];
  }
  __syncthreads();
  if (!active) return;

  v8f acc[4];
#pragma unroll
  for (int t = 0; t < 4; ++t) acc[t] = (v8f)0.0f;

  for (int k = 0; k < KK_; ++k) {
    const int idx = nlds[wave][m * KK_ + k];
    // A fragment: lanes 0-15 hold K={0,1}, lanes 16-31 hold K={2,3} of row m
    v2f a = *(const v2f*)(x + (size_t)idx * CIN_ + 2 * hi);
    const float* wk = &wlds[k * 256 + hi * 128];
#pragma unroll
    for (int t = 0; t < 4; ++t) {
      const int col = t * 16 + m;
      v2f b = *(const v2f*)&wk[col * 2];
      acc[t] = wmma_f32_k4(a, b, acc[t]);
    }
  }

  const int n0 = tile * 16;
#pragma unroll
  for (int t = 0; t < 4; ++t) {
#pragma unroll
    for (int v = 0; v < 8; ++v) {
      out[(size_t)(n0 + v + 8 * hi) * CC_ + t * 16 + m] = acc[t][v];
    }
  }
}

// ---------------------------------------------------------------------------
// Generic 64->64 conv (optionally two weight sets sharing one gather).
// src: [nsrc][64], neigh: [nout][27], Wa/Wb: [27][64][64].
// ---------------------------------------------------------------------------
#define CV_WAVES 4
__global__ __launch_bounds__(32 * CV_WAVES)
void conv64_wmma(const float* __restrict__ src,
                 const int* __restrict__ neigh,
                 const float* __restrict__ Wa,
                 const float* __restrict__ Wb,   // nullptr -> single conv
                 float* __restrict__ outa,
                 float* __restrict__ outb,
                 int nout) {
  // interleaved: wlds[d][pair*128 + col*2 + parity], row = 2*pair+parity
  __shared__ float wlds[2][CC_ * CC_];            // 32 KB
  __shared__ float alds[CV_WAVES][16 * CC_];      // 16 KB
  __shared__ int nlds[CV_WAVES][16 * KK_];        // ~6.9 KB

  const int tid  = threadIdx.x;
  const int wave = tid >> 5;
  const int lane = tid & 31;
  const int m    = lane & 15;
  const int hi   = lane >> 4;

  const bool dual = (Wb != nullptr);
  const int numTiles = nout / 16;
  const int tile = blockIdx.x * CV_WAVES + wave;
  const bool active = tile < numTiles;

  if (active) {
    const int base = tile * 16 * KK_;
    for (int i = lane; i < 16 * KK_; i += 32) nlds[wave][i] = neigh[base + i];
  }

  v8f acc[2][4];
#pragma unroll
  for (int d = 0; d < 2; ++d)
#pragma unroll
    for (int t = 0; t < 4; ++t) acc[d][t] = (v8f)0.0f;

  for (int k = 0; k < KK_; ++k) {
    __syncthreads();  // drains prior iteration's LDS reads (DScnt) first
    // stage weight slice(s), pair-interleaved (block cooperative)
    {
      const float* wa = Wa + (size_t)k * CC_ * CC_;
      for (int i = tid; i < 32 * CC_; i += blockDim.x) {
        const int j = i >> 6;   // row pair 0..31
        const int c = i & 63;
        v2f v;
        v.x = wa[(2 * j) * CC_ + c];
        v.y = wa[(2 * j + 1) * CC_ + c];
        *(v2f*)&wlds[0][j * 128 + c * 2] = v;
      }
      if (dual) {
        const float* wb = Wb + (size_t)k * CC_ * CC_;
        for (int i = tid; i < 32 * CC_; i += blockDim.x) {
          const int j = i >> 6;
          const int c = i & 63;
          v2f v;
          v.x = wb[(2 * j) * CC_ + c];
          v.y = wb[(2 * j + 1) * CC_ + c];
          *(v2f*)&wlds[1][j * 128 + c * 2] = v;
        }
      }
      // prefetch next k's weight slices (one 128B line per thread)
      if (k + 1 < KK_) {
        __builtin_prefetch(Wa + (size_t)(k + 1) * CC_ * CC_ + tid * 32, 0, 3);
        if (dual)
          __builtin_prefetch(Wb + (size_t)(k + 1) * CC_ * CC_ + tid * 32, 0, 3);
      }
    }
    // stage gathered activation rows for this wave's 16 nodes
    if (active) {
      for (int mm = 0; mm < 16; ++mm) {
        const int idx = nlds[wave][mm * KK_ + k];
        const float* gp = src + (size_t)idx * CC_ + lane * 2;
        float* lp = &alds[wave][mm * CC_ + lane * 2];
#ifdef USE_ASYNC_LDS
        __builtin_amdgcn_global_load_async_to_lds_b64(
            (__attribute__((address_space(1))) v2i*)gp,
            (__attribute__((address_space(3))) v2i*)lp, 0, 0);
#else
        *(v2f*)lp = *(const v2f*)gp;
#endif
      }
    }
    __syncthreads();
#ifdef USE_ASYNC_LDS
    if (active) __builtin_amdgcn_s_wait_asynccnt(0);
#endif
    if (!active) continue;

#pragma unroll 4
    for (int s = 0; s < 16; ++s) {
      v2f a = *(const v2f*)&alds[wave][m * CC_ + 4 * s + 2 * hi];
      const float* w0 = &wlds[0][(2 * s + hi) * 128];
#pragma unroll
      for (int t = 0; t < 4; ++t) {
        const int col = t * 16 + m;
        v2f b = *(const v2f*)&w0[col * 2];
        acc[0][t] = wmma_f32_k4(a, b, acc[0][t]);
      }
      if (dual) {
        const float* w1 = &wlds[1][(2 * s + hi) * 128];
#pragma unroll
        for (int t = 0; t < 4; ++t) {
          const int col = t * 16 + m;
          v2f b = *(const v2f*)&w1[col * 2];
          acc[1][t] = wmma_f32_k4(a, b, acc[1][t]);
        }
      }
    }
  }

  if (!active) return;
  const int n0 = tile * 16;
#pragma unroll
  for (int t = 0; t < 4; ++t)
#pragma unroll
    for (int v = 0; v < 8; ++v)
      outa[(size_t)(n0 + v + 8 * hi) * CC_ + t * 16 + m] = acc[0][t][v];
  if (dual) {
#pragma unroll
    for (int t = 0; t < 4; ++t)
#pragma unroll
      for (int v = 0; v < 8; ++v)
        outb[(size_t)(n0 + v + 8 * hi) * CC_ + t * 16 + m] = acc[1][t][v];
  }
}

// ---------------------------------------------------------------------------
// BatchNorm statistics: deterministic two-phase column sum / sum-of-squares.
// ---------------------------------------------------------------------------
#define SUM_ROWS 4096
__global__ __launch_bounds__(256)
void colsum_part(const float* __restrict__ src, int nrows,
                 float* __restrict__ part) {  // [gridDim][128]
  __shared__ float s1[256];
  __shared__ float s2[256];
  const int c   = threadIdx.x & 63;
  const int sub = threadIdx.x >> 6;  // 0..3
  const long r0 = (long)blockIdx.x * SUM_ROWS;
  long rend = r0 + SUM_ROWS;
  if (rend > nrows) rend = nrows;
  float s = 0.f, q = 0.f;
  for (long r = r0 + sub; r < rend; r += 4) {
    float v = src[r * CC_ + c];
    s += v;
    q += v * v;
  }
  s1[threadIdx.x] = s;
  s2[threadIdx.x] = q;
  __syncthreads();
  if (sub == 0) {
    s = s1[c] + s1[64 + c] + s1[128 + c] + s1[192 + c];
    q = s2[c] + s2[64 + c] + s2[128 + c] + s2[192 + c];
    part[(size_t)blockIdx.x * 128 + c] = s;
    part[(size_t)blockIdx.x * 128 + 64 + c] = q;
  }
}

__global__ __launch_bounds__(64)
void bn_params(const float* __restrict__ part, int nchunks, float nrows,
               const float* __restrict__ gamma, const float* __restrict__ beta,
               float* __restrict__ p) {  // p[0..63]=scale, p[64..127]=shift
  const int c = threadIdx.x;
  float s = 0.f, q = 0.f;
  for (int i = 0; i < nchunks; ++i) {
    s += part[(size_t)i * 128 + c];
    q += part[(size_t)i * 128 + 64 + c];
  }
  const float mean = s / nrows;
  const float var  = q / nrows - mean * mean;
  const float sc   = gamma[c] * rsqrtf(var + BN_EPS);
  p[c]      = sc;
  p[64 + c] = beta[c] - mean * sc;
}

// ---------------------------------------------------------------------------
// Octree max-pool fused with BN+ReLU of the stem output.
// max_j relu(bn(x_j)) == relu(max_j bn(x_j)) since relu is monotonic.
// ---------------------------------------------------------------------------
__global__ __launch_bounds__(256)
void pool_bn_relu(const float* __restrict__ raw, const int* __restrict__ child,
                  const float* __restrict__ p, float* __restrict__ out) {
  const int t = blockIdx.x * blockDim.x + threadIdx.x;
  if (t >= N2_ * CC_) return;
  const int c = t & 63;
  const int n = t >> 6;
  const float sc = p[c], sh = p[64 + c];
  float mx = -3.4e38f;
#pragma unroll
  for (int j = 0; j < 8; ++j) {
    const int idx = child[n * 8 + j];
    mx = fmaxf(mx, sc * raw[(size_t)idx * CC_ + c] + sh);
  }
  out[t] = fmaxf(mx, 0.f);
}

__global__ __launch_bounds__(256)
void bn_relu_apply(const float* __restrict__ raw, const float* __restrict__ p,
                   float* __restrict__ out, int n) {
  const int t = blockIdx.x * blockDim.x + threadIdx.x;
  if (t >= n) return;
  const int c = t & 63;
  out[t] = fmaxf(p[c] * raw[t] + p[64 + c], 0.f);
}

__global__ __launch_bounds__(256)
void final_add_relu(const float* __restrict__ raw2, const float* __restrict__ rawd,
                    const float* __restrict__ p2, const float* __restrict__ pd,
                    float* __restrict__ out, int n) {
  const int t = blockIdx.x * blockDim.x + threadIdx.x;
  if (t >= n) return;
  const int c = t & 63;
  const float a = p2[c] * raw2[t] + p2[64 + c];
  const float b = pd[c] * rawd[t] + pd[64 + c];
  out[t] = fmaxf(a + b, 0.f);
}

// ---------------------------------------------------------------------------
extern "C" void kernel_launch(void* const* d_in, const int* in_sizes, int n_in,
                              void* d_out, int out_size, void* d_ws, size_t ws_size,
                              hipStream_t stream) {
  const float* x          = (const float*)d_in[0];
  const int*   neigh_stem = (const int*)d_in[1];
  const int*   pool_child = (const int*)d_in[2];
  const int*   neigh_ds   = (const int*)d_in[3];
  const int*   neigh_res  = (const int*)d_in[4];
  const float* W_stem     = (const float*)d_in[5];
  const float* g_stem     = (const float*)d_in[6];
  const float* b_stem     = (const float*)d_in[7];
  const float* W1         = (const float*)d_in[8];
  const float* g1         = (const float*)d_in[9];
  const float* b1         = (const float*)d_in[10];
  const float* W2         = (const float*)d_in[11];
  const float* g2         = (const float*)d_in[12];
  const float* b2         = (const float*)d_in[13];
  const float* Wd         = (const float*)d_in[14];
  const float* gd         = (const float*)d_in[15];
  const float* bd         = (const float*)d_in[16];
  float* out = (float*)d_out;

  // workspace layout
  float* ws    = (float*)d_ws;
  float* raw1  = ws;                           // N1*64
  float* h2    = raw1 + (size_t)N1_ * CC_;     // N2*64
  float* rc1   = h2   + (size_t)N2_ * CC_;     // N3*64
  float* rcd   = rc1  + (size_t)N3_ * CC_;     // N3*64
  float* o1n   = rcd  + (size_t)N3_ * CC_;     // N3*64
  float* rc2   = o1n  + (size_t)N3_ * CC_;     // N3*64
  float* partA = rc2  + (size_t)N3_ * CC_;     // <=80*128 each
  float* partB = partA + 128 * 80;
  float* partC = partB + 128 * 80;
  float* partD = partC + 128 * 80;
  float* pS    = partD + 128 * 80;
  float* p1    = pS + 128;
  float* pd    = p1 + 128;
  float* p2    = pd + 128;

  const int chunks1 = (N1_ + SUM_ROWS - 1) / SUM_ROWS;  // 74
  const int chunks3 = (N3_ + SUM_ROWS - 1) / SUM_ROWS;  // 8

  // 1) stem conv
  {
    const int tiles = N1_ / 16;
    const int blocks = (tiles + STEM_WAVES - 1) / STEM_WAVES;
    stem_conv_wmma<<<blocks, 32 * STEM_WAVES, 0, stream>>>(x, neigh_stem, W_stem, raw1);
  }
  // 2) BN stats for stem
  colsum_part<<<chunks1, 256, 0, stream>>>(raw1, N1_, partA);
  bn_params<<<1, 64, 0, stream>>>(partA, chunks1, (float)N1_, g_stem, b_stem, pS);
  // 3) BN+ReLU+maxpool -> h2
  pool_bn_relu<<<(N2_ * CC_) / 256, 256, 0, stream>>>(raw1, pool_child, pS, h2);
  // 4) conv1 + downsample conv fused (shared gather over neigh_ds)
  {
    const int tiles = N3_ / 16;
    const int blocks = (tiles + CV_WAVES - 1) / CV_WAVES;
    conv64_wmma<<<blocks, 32 * CV_WAVES, 0, stream>>>(h2, neigh_ds, W1, Wd, rc1, rcd, N3_);
  }
  // 5) BN stats + apply BN/ReLU to conv1
  colsum_part<<<chunks3, 256, 0, stream>>>(rc1, N3_, partB);
  colsum_part<<<chunks3, 256, 0, stream>>>(rcd, N3_, partC);
  bn_params<<<1, 64, 0, stream>>>(partB, chunks3, (float)N3_, g1, b1, p1);
  bn_params<<<1, 64, 0, stream>>>(partC, chunks3, (float)N3_, gd, bd, pd);
  bn_relu_apply<<<(N3_ * CC_) / 256, 256, 0, stream>>>(rc1, p1, o1n, N3_ * CC_);
  // 6) conv2
  {
    const int tiles = N3_ / 16;
    const int blocks = (tiles + CV_WAVES - 1) / CV_WAVES;
    conv64_wmma<<<blocks, 32 * CV_WAVES, 0, stream>>>(o1n, neigh_res, W2, nullptr, rc2, nullptr, N3_);
  }
  // 7) BN stats for conv2, then fused residual add + ReLU
  colsum_part<<<chunks3, 256, 0, stream>>>(rc2, N3_, partD);
  bn_params<<<1, 64, 0, stream>>>(partD, chunks3, (float)N3_, g2, b2, p2);
  final_add_relu<<<(N3_ * CC_) / 256, 256, 0, stream>>>(rc2, rcd, p2, pd, out, N3_ * CC_);
}